// Router_11665131176297
// MI455X (gfx1250) — compile-verified
//
#include <hip/hip_runtime.h>
#include <hip/hip_bf16.h>
#include <math.h>

typedef __attribute__((ext_vector_type(2))) float v2f;
typedef __attribute__((ext_vector_type(8))) float v8f;

#define WAVES_PER_BLOCK 8
#define TOKENS_PER_WAVE 16
#define NEXPERTS 64

// ---------------------------------------------------------------------------
// Workspace layout (floats): ws[0..63] = per-expert routing-weight sums over T
//                            ws[64]    = sum of logits^2 over T*E
// ---------------------------------------------------------------------------

__global__ void router_init_ws(float* __restrict__ ws) {
    int i = threadIdx.x;
    if (i < NEXPERTS + 1) ws[i] = 0.0f;
}

__global__ __launch_bounds__(WAVES_PER_BLOCK * 32)
void router_main_kernel(const float* __restrict__ x,
                        const float* __restrict__ Wg,
                        const float* __restrict__ gamma,
                        const float* __restrict__ beta,
                        const float* __restrict__ temperature,
                        float* __restrict__ out,
                        float* __restrict__ ws,
                        int D, long long T) {
    __shared__ float lds[WAVES_PER_BLOCK * TOKENS_PER_WAVE * NEXPERTS]; // 32 KB
    __shared__ float ssbuf[WAVES_PER_BLOCK * TOKENS_PER_WAVE];          // 512 B

    const int lane = threadIdx.x & 31;
    const int wave = threadIdx.x >> 5;
    const long long tokenBase =
        ((long long)blockIdx.x * WAVES_PER_BLOCK + wave) * TOKENS_PER_WAVE;

    const int r    = lane & 15;        // row within 16 (token / expert index)
    const int koff = (lane >> 4) << 1; // K pair select: lanes 0-15 -> K{0,1}, 16-31 -> K{2,3}

    // A fragment stream: 16 tokens x K  (fp32 16x16x4 layout)
    const float* __restrict__ xA = x + (tokenBase + r) * (long long)D + koff;
    // B fragment streams: 4 expert tiles of 16, B[k][n] = W[n][k]
    const float* __restrict__ wB = Wg + (long long)r * D + koff;
    const long long tstride = 16ll * D;

    v8f acc0 = {}, acc1 = {}, acc2 = {}, acc3 = {};

#pragma unroll 4
    for (int k = 0; k < D; k += 4) {
        v2f a  = *(const v2f*)(xA + k);
        v2f b0 = *(const v2f*)(wB + k);
        v2f b1 = *(const v2f*)(wB + tstride + k);
        v2f b2 = *(const v2f*)(wB + 2 * tstride + k);
        v2f b3 = *(const v2f*)(wB + 3 * tstride + k);
        acc0 = __builtin_amdgcn_wmma_f32_16x16x4_f32(false, a, false, b0, (short)0, acc0, false, false);
        acc1 = __builtin_amdgcn_wmma_f32_16x16x4_f32(false, a, false, b1, (short)0, acc1, false, false);
        acc2 = __builtin_amdgcn_wmma_f32_16x16x4_f32(false, a, false, b2, (short)0, acc2, false, false);
        acc3 = __builtin_amdgcn_wmma_f32_16x16x4_f32(false, a, false, b3, (short)0, acc3, false, false);
    }

    // --- spill z tile to per-wave LDS slab, per C/D layout:
    //     lanes 0-15: N=lane, M=vgpr ; lanes 16-31: N=lane-16, M=8+vgpr
    float* slab = &lds[wave * TOKENS_PER_WAVE * NEXPERTS];
    const int trow = (lane >> 4) << 3; // 0 or 8
    const int ncol = lane & 15;
#pragma unroll
    for (int m = 0; m < 8; ++m) {
        slab[(trow + m) * NEXPERTS +  0 + ncol] = acc0[m];
        slab[(trow + m) * NEXPERTS + 16 + ncol] = acc1[m];
        slab[(trow + m) * NEXPERTS + 32 + ncol] = acc2[m];
        slab[(trow + m) * NEXPERTS + 48 + ncol] = acc3[m];
    }
    __syncthreads();

    const long long TE = T * (long long)NEXPERTS;
    const float invTemp = 1.0f / (fabsf(temperature[0]) + 1e-6f);

    if (lane < TOKENS_PER_WAVE) {
        float* zr = &slab[lane * NEXPERTS];
        // LayerNorm over experts
        float mu = 0.0f;
#pragma unroll
        for (int e = 0; e < NEXPERTS; ++e) mu += zr[e];
        mu *= (1.0f / NEXPERTS);
        float var = 0.0f;
#pragma unroll
        for (int e = 0; e < NEXPERTS; ++e) { float d = zr[e] - mu; var += d * d; }
        var *= (1.0f / NEXPERTS);
        const float rstd = rsqrtf(var + 1e-5f);

        // logits (overwrite in LDS), track max + sum of squares (z-loss)
        float mx = -3.4e38f, ss = 0.0f;
#pragma unroll
        for (int e = 0; e < NEXPERTS; ++e) {
            float l = ((zr[e] - mu) * rstd * gamma[e] + beta[e]) * invTemp;
            zr[e] = l;
            ss += l * l;
            mx = fmaxf(mx, l);
        }
        // softmax
        float sum = 0.0f;
#pragma unroll
        for (int e = 0; e < NEXPERTS; ++e) { float w = __expf(zr[e] - mx); zr[e] = w; sum += w; }
        const float inv = 1.0f / sum;

        // normalize + top-2 (first occurrence wins ties, like lax.top_k)
        float w1 = -1.0f, w2 = -1.0f; int i1 = 0, i2 = 0;
#pragma unroll
        for (int e = 0; e < NEXPERTS; ++e) {
            float w = zr[e] * inv;
            zr[e] = w;
            if (w > w1)      { w2 = w1; i2 = i1; w1 = w; i1 = e; }
            else if (w > w2) { w2 = w;  i2 = e; }
        }

        const long long token = tokenBase + lane;
        float* __restrict__ rw = out + token * NEXPERTS;
        float* __restrict__ dm = out + TE + token * NEXPERTS;
        const float s2 = 1.0f / (w1 + w2 + 1e-6f);
#pragma unroll
        for (int e = 0; e < NEXPERTS; ++e) { rw[e] = zr[e]; dm[e] = 0.0f; }
        dm[i1] = w1 * s2;
        dm[i2] = w2 * s2;

        ssbuf[wave * TOKENS_PER_WAVE + lane] = ss;
    }
    __syncthreads();

    // per-wave expert-load partials -> one atomic per expert per wave
    {
        float sa = 0.0f, sb = 0.0f;
#pragma unroll
        for (int t = 0; t < TOKENS_PER_WAVE; ++t) {
            sa += slab[t * NEXPERTS + lane];
            sb += slab[t * NEXPERTS + lane + 32];
        }
        atomicAdd(&ws[lane], sa);
        atomicAdd(&ws[lane + 32], sb);
    }
    if (lane == 0) {
        float s = 0.0f;
#pragma unroll
        for (int t = 0; t < TOKENS_PER_WAVE; ++t) s += ssbuf[wave * TOKENS_PER_WAVE + t];
        atomicAdd(&ws[NEXPERTS], s);
    }
}

__global__ void router_finalize(const float* __restrict__ ws,
                                float* __restrict__ out,
                                int D, long long T) {
    if (threadIdx.x == 0 && blockIdx.x == 0) {
        const long long TE = T * (long long)NEXPERTS;
        const float z_loss = ws[NEXPERTS] / (float)TE;
        const float ideal = 1.0f / (float)NEXPERTS;
        const float logIdeal = logf(ideal);
        float kl = 0.0f;
        for (int e = 0; e < NEXPERTS; ++e) {
            float actual = ws[e] / (float)T;
            kl += ideal * (logIdeal - logf(actual));
        }
        kl *= (1.0f / (float)NEXPERTS);
        out[2 * TE] = 0.01f * z_loss + 0.01f * kl;
    }
}

extern "C" void kernel_launch(void* const* d_in, const int* in_sizes, int n_in,
                              void* d_out, int out_size, void* d_ws, size_t ws_size,
                              hipStream_t stream) {
    const float* x     = (const float*)d_in[0];
    const float* Wg    = (const float*)d_in[1];
    const float* gamma = (const float*)d_in[2];
    const float* beta  = (const float*)d_in[3];
    const float* temp  = (const float*)d_in[4];
    float* out = (float*)d_out;
    float* ws  = (float*)d_ws;

    const int E = in_sizes[2];                       // 64
    const int D = in_sizes[1] / E;                   // 4096
    const long long T = (long long)in_sizes[0] / D;  // 32768

    router_init_ws<<<1, 128, 0, stream>>>(ws);

    const int tokensPerBlock = WAVES_PER_BLOCK * TOKENS_PER_WAVE; // 128
    const int nblocks = (int)((T + tokensPerBlock - 1) / tokensPerBlock);
    router_main_kernel<<<nblocks, WAVES_PER_BLOCK * 32, 0, stream>>>(
        x, Wg, gamma, beta, temp, out, ws, D, T);

    router_finalize<<<1, 64, 0, stream>>>(ws, out, D, T);
}